// xLSTMModel_86500641341761
// MI455X (gfx1250) — compile-verified
//
#include <hip/hip_runtime.h>
#include <math.h>

// ---------------------------------------------------------------------------
// xLSTM forward for MI455X (gfx1250, wave32).
// All heavy matmuls use V_WMMA_F32_16X16X4_F32 (fp32-exact WMMA path).
// GEMM: one wave computes a 16x64 output strip (4 accumulators) for 4x
// A-fragment reuse -> higher arithmetic intensity vs L2.
// ---------------------------------------------------------------------------

typedef __attribute__((ext_vector_type(2))) float v2f;
typedef __attribute__((ext_vector_type(8))) float v8f;

#define WMMA_F32(a, b, c) \
  __builtin_amdgcn_wmma_f32_16x16x4_f32(false, (a), false, (b), (short)0, (c), false, false)

__device__ __forceinline__ float siluf(float x) { return x / (1.f + expf(-x)); }
__device__ __forceinline__ float logsigf(float x) {
  return (x >= 0.f) ? -log1pf(expf(-x)) : (x - log1pf(expf(x)));
}
__device__ __forceinline__ float geluf(float x) {
  return 0.5f * x * (1.f + erff(x * 0.70710678118654752f));
}

// ------------------------ generic fp32 WMMA GEMM ----------------------------
// C = A(MxK,row) @ B(KxN,row) [+ Cin].  One wave per 16x64 strip (N%64==0),
// 4 waves per block.
__global__ void __launch_bounds__(128)
gemm_f32_wmma(const float* __restrict__ A, const float* __restrict__ B,
              const float* __restrict__ Cin, float* __restrict__ Cout,
              int M, int N, int K, int tiles)
{
  int wave = blockIdx.x * (blockDim.x >> 5) + (threadIdx.x >> 5);
  if (wave >= tiles) return;                      // wave-uniform branch
  int lane = threadIdx.x & 31;
  int tilesN = N >> 6;                            // 64-wide strips
  int tM = wave / tilesN, tN = wave - tM * tilesN;
  int row0 = tM << 4, col0 = tN << 6;
  int hi = lane >> 4, lo = lane & 15;

  v8f acc0 = {}, acc1 = {}, acc2 = {}, acc3 = {};
  const float* Ap = A + (size_t)(row0 + lo) * K + 2 * hi;
  const float* Bp = B + (size_t)(2 * hi) * N + col0 + lo;
  for (int k = 0; k < K; k += 4) {
    v2f af;
    af.x = Ap[k];
    af.y = Ap[k + 1];
    const float* bp0 = Bp + (size_t)k * N;
    const float* bp1 = Bp + (size_t)(k + 1) * N;
    v2f bf;
    bf.x = bp0[0];  bf.y = bp1[0];  acc0 = WMMA_F32(af, bf, acc0);
    bf.x = bp0[16]; bf.y = bp1[16]; acc1 = WMMA_F32(af, bf, acc1);
    bf.x = bp0[32]; bf.y = bp1[32]; acc2 = WMMA_F32(af, bf, acc2);
    bf.x = bp0[48]; bf.y = bp1[48]; acc3 = WMMA_F32(af, bf, acc3);
  }
#pragma unroll
  for (int r = 0; r < 8; ++r) {
    size_t idx = (size_t)(row0 + r + 8 * hi) * N + col0 + lo;
    float v0 = acc0[r], v1 = acc1[r], v2 = acc2[r], v3 = acc3[r];
    if (Cin) {
      v0 += Cin[idx];
      v1 += Cin[idx + 16];
      v2 += Cin[idx + 32];
      v3 += Cin[idx + 48];
    }
    Cout[idx]      = v0;
    Cout[idx + 16] = v1;
    Cout[idx + 32] = v2;
    Cout[idx + 48] = v3;
  }
}

// ------------------------------- layernorm ---------------------------------
// one 128-thread block per row of width 128
__global__ void layernorm128(const float* __restrict__ in, const float* __restrict__ w,
                             float* __restrict__ out)
{
  __shared__ float red[128];
  int r = blockIdx.x, t = threadIdx.x;
  float v = in[(size_t)r * 128 + t];
  red[t] = v; __syncthreads();
  for (int o = 64; o; o >>= 1) { if (t < o) red[t] += red[t + o]; __syncthreads(); }
  float mu = red[0] * (1.f / 128.f);
  __syncthreads();
  float d = v - mu;
  red[t] = d * d; __syncthreads();
  for (int o = 64; o; o >>= 1) { if (t < o) red[t] += red[t + o]; __syncthreads(); }
  float var = red[0] * (1.f / 128.f);
  out[(size_t)r * 128 + t] = d * rsqrtf(var + 1e-5f) * w[t];
}

// --------------------------- causal conv1d + silu ---------------------------
// in: (B,S,ld_in) channel-last (uses first C channels), out: (B,S,C)
__global__ void conv_silu(const float* __restrict__ in, int ld_in,
                          const float* __restrict__ W, const float* __restrict__ bias,
                          float* __restrict__ out, int C, int total)
{
  int idx = blockIdx.x * blockDim.x + threadIdx.x;
  if (idx >= total) return;
  int c = idx % C;
  int s = (idx / C) & 511;
  int b = idx / (C * 512);
  float acc = bias[c];
#pragma unroll
  for (int k = 0; k < 4; ++k) {
    int ss = s - 3 + k;
    if (ss >= 0) acc += W[c * 4 + k] * in[((size_t)b * 512 + ss) * ld_in + c];
  }
  out[((size_t)b * 512 + s) * C + c] = siluf(acc);
}

// ------------------------- headwise 4x4 projection --------------------------
__global__ void headwise4(const float* __restrict__ x, int ld_x,
                          const float* __restrict__ W, float* __restrict__ out, int total)
{
  int idx = blockIdx.x * blockDim.x + threadIdx.x;
  if (idx >= total) return;
  int c = idx & 255;
  int bs = idx >> 8;
  int n = c >> 2, o = c & 3;
  const float* xp = x + (size_t)bs * ld_x + n * 4;
  const float* wp = W + (n * 4 + o) * 4;
  out[(size_t)bs * 256 + c] = xp[0] * wp[0] + xp[1] * wp[1] + xp[2] * wp[2] + xp[3] * wp[3];
}

// -------------------- ig / fg gate projections (dot-768) --------------------
// 8 wave-groups per block: groups 0..3 -> ig heads, 4..7 -> fg heads
__global__ void mlstm_gates(const float* __restrict__ Q, const float* __restrict__ Kb,
                            const float* __restrict__ V,
                            const float* __restrict__ igW, const float* __restrict__ igb,
                            const float* __restrict__ fgW, const float* __restrict__ fgb,
                            float* __restrict__ ig, float* __restrict__ fg)
{
  int bs = blockIdx.x;
  int b = bs >> 9, s = bs & 511;
  int t = threadIdx.x, lane = t & 31, grp = t >> 5;
  int h = grp & 3;
  bool isF = grp >= 4;
  const float* W = (isF ? fgW : igW) + h * 768;
  float acc = 0.f;
  for (int j = lane; j < 768; j += 32) {
    float xv;
    if (j < 256)      xv = Q[(size_t)bs * 256 + j];
    else if (j < 512) xv = Kb[(size_t)bs * 256 + j - 256];
    else              xv = V[(size_t)bs * 256 + j - 512];
    acc += xv * W[j];
  }
  for (int m = 16; m; m >>= 1) acc += __shfl_xor(acc, m);
  if (lane == 0) {
    float* dst = isF ? fg : ig;
    dst[((size_t)b * 4 + h) * 512 + s] = acc + (isF ? fgb : igb)[h];
  }
}

// ------------- per-(b,h) scans: csum = prefix-sum(logsig(fg)),  -------------
// ------------- a = ig - csum, amax = prefix-max(a)              -------------
__global__ void gate_scan(const float* __restrict__ fg, const float* __restrict__ ig,
                          float* __restrict__ csum, float* __restrict__ a,
                          float* __restrict__ amax)
{
  __shared__ float buf[512];
  int bh = blockIdx.x, t = threadIdx.x;
  size_t base = (size_t)bh * 512;
  float lsf = logsigf(fg[base + t]);
  buf[t] = lsf; __syncthreads();
  for (int off = 1; off < 512; off <<= 1) {
    float v = buf[t];
    float w = (t >= off) ? buf[t - off] : 0.f;
    __syncthreads();
    buf[t] = v + w;
    __syncthreads();
  }
  float cs = buf[t];
  csum[base + t] = cs;
  float av = ig[base + t] - cs;
  a[base + t] = av;
  __syncthreads();
  buf[t] = av; __syncthreads();
  for (int off = 1; off < 512; off <<= 1) {
    float v = buf[t];
    float w = (t >= off) ? buf[t - off] : -3.4e38f;
    __syncthreads();
    buf[t] = fmaxf(v, w);
    __syncthreads();
  }
  amax[base + t] = buf[t];
}

// ---------------------- mLSTM attention (WMMA, tiled) -----------------------
// grid = (S/16, NH, B), block = 32 (one wave)
__global__ void __launch_bounds__(32)
mlstm_attn(const float* __restrict__ Q, const float* __restrict__ Kb,
           const float* __restrict__ V,
           const float* __restrict__ aV, const float* __restrict__ amaxV,
           const float* __restrict__ csumV, float* __restrict__ H)
{
  __shared__ float ct[16][17];
  int qt = blockIdx.x, h = blockIdx.y, b = blockIdx.z;
  int lane = threadIdx.x, hi = lane >> 4, lo = lane & 15;
  int sbase = qt * 16;
  size_t bhS = ((size_t)b * 4 + h) * 512;

  float amax_r[8], nexp_r[8];
#pragma unroll
  for (int r = 0; r < 8; ++r) {
    int s = sbase + r + 8 * hi;
    amax_r[r] = amaxV[bhS + s];
    nexp_r[r] = expf(-csumV[bhS + s] - amax_r[r]);
  }

  v8f acc0 = {}, acc1 = {}, acc2 = {}, acc3 = {}, acc4 = {};

  for (int tb = 0; tb <= sbase; tb += 16) {
    // ---- S_tile = Q Kt  (16x16, K=64) ----
    v8f qk = {};
    const float* qp = Q + ((size_t)b * 512 + sbase + lo) * 256 + h * 64 + 2 * hi;
    const float* kp = Kb + ((size_t)b * 512 + tb + lo) * 256 + h * 64 + 2 * hi;
#pragma unroll
    for (int kk = 0; kk < 64; kk += 4) {
      v2f af, bf;
      af.x = qp[kk]; af.y = qp[kk + 1];
      bf.x = kp[kk]; bf.y = kp[kk + 1];
      qk = WMMA_F32(af, bf, qk);
    }
    // ---- weight: C = qk * DH^-0.5 * exp(a[t]-amax[s]) with causal mask ----
    float at = aV[bhS + tb + lo];
    int tcol = tb + lo;
#pragma unroll
    for (int r = 0; r < 8; ++r) {
      int s = sbase + r + 8 * hi;
      float w = (tcol <= s) ? 0.125f * expf(at - amax_r[r]) : 0.f;
      ct[r + 8 * hi][lo] = qk[r] * w;
    }
    __syncthreads();
    // ---- PV: acc[n] += C @ V_tile ; acc4 += C @ ones (row sums) ----
#pragma unroll
    for (int k = 0; k < 16; k += 4) {
      v2f af;
      af.x = ct[lo][k + 2 * hi];
      af.y = ct[lo][k + 2 * hi + 1];
      const float* vp = V + ((size_t)b * 512 + tb + k + 2 * hi) * 256 + h * 64 + lo;
      v2f bf;
      bf.x = vp[0];  bf.y = vp[256];  acc0 = WMMA_F32(af, bf, acc0);
      bf.x = vp[16]; bf.y = vp[272];  acc1 = WMMA_F32(af, bf, acc1);
      bf.x = vp[32]; bf.y = vp[288];  acc2 = WMMA_F32(af, bf, acc2);
      bf.x = vp[48]; bf.y = vp[304];  acc3 = WMMA_F32(af, bf, acc3);
      float one = (lo == 0) ? 1.f : 0.f;
      v2f of; of.x = one; of.y = one;
      acc4 = WMMA_F32(af, of, acc4);
    }
    __syncthreads();
  }

  // ---- finalize: h = num / (max(|rowsum|, exp(-maxlogD)) + 1e-6) ----
#pragma unroll
  for (int r = 0; r < 8; ++r) {
    float rs = __shfl(acc4[r], lane & 16);  // broadcast column 0 of this half
    float inv = 1.f / (fmaxf(fabsf(rs), nexp_r[r]) + 1e-6f);
    size_t orow = ((size_t)b * 512 + sbase + r + 8 * hi) * 256 + h * 64 + lo;
    H[orow]      = acc0[r] * inv;
    H[orow + 16] = acc1[r] * inv;
    H[orow + 32] = acc2[r] * inv;
    H[orow + 48] = acc3[r] * inv;
  }
}

// ------------------------- multihead group norm -----------------------------
// one 32-lane wave per (bs, head); DHh in {32,64}; safe in-place
__global__ void groupnorm_ws(const float* __restrict__ in, const float* __restrict__ w,
                             float* __restrict__ out, int NHh, int DHh, int groups)
{
  int gid = (blockIdx.x * blockDim.x + threadIdx.x) >> 5;
  int lane = threadIdx.x & 31;
  if (gid >= groups) return;
  int h = gid % NHh;
  int bs = gid / NHh;
  const float* ip = in + (size_t)bs * NHh * DHh + h * DHh;
  float v0 = ip[lane];
  float v1 = (DHh == 64) ? ip[lane + 32] : 0.f;
  float sum = v0 + v1, sq = v0 * v0 + v1 * v1;
  for (int m = 16; m; m >>= 1) { sum += __shfl_xor(sum, m); sq += __shfl_xor(sq, m); }
  float invn = 1.f / (float)DHh;
  float mu = sum * invn;
  float var = sq * invn - mu * mu;
  float inv = rsqrtf(var + 1e-5f);
  float* op = out + (size_t)bs * NHh * DHh + h * DHh;
  const float* wp = w + h * DHh;
  op[lane] = (v0 - mu) * inv * wp[lane];
  if (DHh == 64) op[lane + 32] = (v1 - mu) * inv * wp[lane + 32];
}

// -------------------- h = (hn + skip*xc) * silu(z) --------------------------
__global__ void mlstm_combine(const float* __restrict__ Hn, const float* __restrict__ XC,
                              const float* __restrict__ UP, const float* __restrict__ skip,
                              float* __restrict__ Hc, int total)
{
  int idx = blockIdx.x * blockDim.x + threadIdx.x;
  if (idx >= total) return;
  int c = idx & 255;
  int bs = idx >> 8;
  float z = UP[(size_t)bs * 512 + 256 + c];
  Hc[idx] = (Hn[idx] + skip[c] * XC[idx]) * siluf(z);
}

// ----------------------- sLSTM gate pre-activations -------------------------
// gx[b,s,n,g,o] = bias[n,g,o] + sum_d src[b,s,n*32+d] * gW[g][n][o][d]
__global__ void slstm_gates(const float* __restrict__ XC, const float* __restrict__ XN,
                            const float* __restrict__ gW, const float* __restrict__ bias,
                            float* __restrict__ gx, int total)
{
  int idx = blockIdx.x * blockDim.x + threadIdx.x;
  if (idx >= total) return;
  int o = idx & 31;
  int g = (idx >> 5) & 3;
  int n = (idx >> 7) & 3;
  int bs = idx >> 9;
  const float* src = (g <= 1) ? XC : XN;
  const float* xp = src + (size_t)bs * 128 + n * 32;
  const float* wp = gW + (((size_t)g * 4 + n) * 32 + o) * 32;
  float acc = bias[(n * 4 + g) * 32 + o];
#pragma unroll
  for (int d = 0; d < 32; ++d) acc += xp[d] * wp[d];
  gx[idx] = acc;
}

// --------------------------- sLSTM recurrence -------------------------------
// one block per (b, head); 128 threads = 4 gates x 32 dims; R cached in LDS
__global__ void __launch_bounds__(128)
slstm_scan(const float* __restrict__ gx, const float* __restrict__ R,
           float* __restrict__ hs)
{
  __shared__ float Rs[32 * 128];
  __shared__ float hsh[32];
  __shared__ float raw[128];
  int b = blockIdx.x >> 2, n = blockIdx.x & 3;
  int t = threadIdx.x, g = t >> 5, d = t & 31;
  for (int i = t; i < 32 * 128; i += 128) Rs[i] = R[(size_t)n * 32 * 128 + i];
  if (t < 32) hsh[t] = 0.f;
  float c = 0.f, nn = 0.f, m = 0.f;
  __syncthreads();
  for (int s = 0; s < 512; ++s) {
    float acc = gx[(((size_t)(b * 512 + s) * 4 + n) * 4 + g) * 32 + d];
#pragma unroll
    for (int dd = 0; dd < 32; ++dd) acc += hsh[dd] * Rs[dd * 128 + g * 32 + d];
    raw[t] = acc;
    __syncthreads();
    if (t < 32) {
      float ir = raw[d], fr = raw[32 + d], zr = raw[64 + d], og = raw[96 + d];
      float lfm = m + logsigf(fr);
      float mn = fmaxf(ir, lfm);
      float iv = expf(ir - mn), fv = expf(lfm - mn);
      c = fv * c + iv * tanhf(zr);
      nn = fv * nn + iv;
      m = mn;
      float hv = (1.f / (1.f + expf(-og))) * c / nn;
      hsh[d] = hv;
      hs[((size_t)(b * 512 + s)) * 128 + n * 32 + d] = hv;
    }
    __syncthreads();
  }
}

// ------------------------------ small elementwise ---------------------------
__global__ void add_inplace(float* __restrict__ X, const float* __restrict__ Y, int total)
{
  int idx = blockIdx.x * blockDim.x + threadIdx.x;
  if (idx < total) X[idx] += Y[idx];
}

__global__ void gelu_gate(const float* __restrict__ FFb, float* __restrict__ GF, int total)
{
  int idx = blockIdx.x * blockDim.x + threadIdx.x;
  if (idx >= total) return;
  int j = idx % 192;
  int bs = idx / 192;
  float pre = FFb[(size_t)bs * 384 + j];
  float up = FFb[(size_t)bs * 384 + 192 + j];
  GF[idx] = geluf(pre) * up;
}

// ---------------------- final FC on last timestep ---------------------------
__global__ void final_fc(const float* __restrict__ XN, const float* __restrict__ fcW,
                         const float* __restrict__ fcb, float* __restrict__ out)
{
  __shared__ float red[128];
  int b = blockIdx.x, t = threadIdx.x;
  red[t] = XN[((size_t)b * 512 + 511) * 128 + t] * fcW[t];
  __syncthreads();
  for (int o = 64; o; o >>= 1) { if (t < o) red[t] += red[t + o]; __syncthreads(); }
  if (t == 0) out[b] = red[0] + fcb[0];
}

// ---------------------------------------------------------------------------
extern "C" void kernel_launch(void* const* d_in, const int* in_sizes, int n_in,
                              void* d_out, int out_size, void* d_ws, size_t ws_size,
                              hipStream_t stream)
{
  (void)in_sizes; (void)n_in; (void)out_size; (void)ws_size;
  const float* x_in       = (const float*)d_in[0];
  const float* m_ln_w     = (const float*)d_in[1];
  const float* m_up_W     = (const float*)d_in[2];
  const float* m_conv_W   = (const float*)d_in[3];
  const float* m_conv_b   = (const float*)d_in[4];
  const float* m_q_W      = (const float*)d_in[5];
  const float* m_k_W      = (const float*)d_in[6];
  const float* m_v_W      = (const float*)d_in[7];
  const float* m_ig_W     = (const float*)d_in[8];
  const float* m_ig_b     = (const float*)d_in[9];
  const float* m_fg_W     = (const float*)d_in[10];
  const float* m_fg_b     = (const float*)d_in[11];
  const float* m_on_w     = (const float*)d_in[12];
  const float* m_skip     = (const float*)d_in[13];
  const float* m_down_W   = (const float*)d_in[14];
  const float* s_ln_w     = (const float*)d_in[15];
  const float* s_conv_W   = (const float*)d_in[16];
  const float* s_conv_b   = (const float*)d_in[17];
  const float* s_gates_W  = (const float*)d_in[18];
  const float* s_R        = (const float*)d_in[19];
  const float* s_bias     = (const float*)d_in[20];
  const float* s_gn_w     = (const float*)d_in[21];
  const float* s_ffn_ln_w = (const float*)d_in[22];
  const float* s_ffn_up_W = (const float*)d_in[23];
  const float* s_ffn_dn_W = (const float*)d_in[24];
  const float* post_norm  = (const float*)d_in[25];
  const float* fc_W       = (const float*)d_in[26];
  const float* fc_b       = (const float*)d_in[27];

  const int BS = 16 * 512;                  // 8192 rows
  float* ws  = (float*)d_ws;
  float* X   = ws;                          // 1,048,576
  float* XN  = X  + 1048576;                // 1,048,576
  float* UP  = XN + 1048576;                // 4,194,304 (also sLSTM gx / FFN buf)
  float* XC  = UP + 4194304;                // 2,097,152
  float* Qb  = XC + 2097152;                // 2,097,152 (also Hc)
  float* Kb  = Qb + 2097152;                // 2,097,152 (also sLSTM hs)
  float* Vb  = Kb + 2097152;                // 2,097,152 (also FFN gated)
  float* Hb  = Vb + 2097152;                // 2,097,152
  float* igB = Hb + 2097152;                // 32,768
  float* fgB = igB + 32768;
  float* csB = fgB + 32768;
  float* aB  = csB + 32768;
  float* amB = aB  + 32768;

  hipMemcpyAsync(X, x_in, sizeof(float) * 1048576, hipMemcpyDeviceToDevice, stream);

  auto gemm = [&](const float* A, const float* Bm, const float* Cin, float* Cout,
                  int M, int N, int K) {
    int tiles = (M / 16) * (N / 64);        // 16x64 strips per wave
    gemm_f32_wmma<<<dim3((tiles + 3) / 4), dim3(128), 0, stream>>>(A, Bm, Cin, Cout, M, N, K, tiles);
  };

  auto mlstm = [&](int mi) {
    layernorm128<<<dim3(BS), dim3(128), 0, stream>>>(X, m_ln_w + mi * 128, XN);
    gemm(XN, m_up_W + (size_t)mi * 128 * 512, nullptr, UP, BS, 512, 128);
    conv_silu<<<dim3(8192), dim3(256), 0, stream>>>(UP, 512, m_conv_W + mi * 1024,
                                                    m_conv_b + mi * 256, XC, 256, BS * 256);
    headwise4<<<dim3(8192), dim3(256), 0, stream>>>(XC, 256, m_q_W + mi * 1024, Qb, BS * 256);
    headwise4<<<dim3(8192), dim3(256), 0, stream>>>(XC, 256, m_k_W + mi * 1024, Kb, BS * 256);
    headwise4<<<dim3(8192), dim3(256), 0, stream>>>(UP, 512, m_v_W + mi * 1024, Vb, BS * 256);
    mlstm_gates<<<dim3(BS), dim3(256), 0, stream>>>(Qb, Kb, Vb,
                                                    m_ig_W + mi * 4 * 768, m_ig_b + mi * 4,
                                                    m_fg_W + mi * 4 * 768, m_fg_b + mi * 4,
                                                    igB, fgB);
    gate_scan<<<dim3(64), dim3(512), 0, stream>>>(fgB, igB, csB, aB, amB);
    mlstm_attn<<<dim3(32, 4, 16), dim3(32), 0, stream>>>(Qb, Kb, Vb, aB, amB, csB, Hb);
    groupnorm_ws<<<dim3(BS), dim3(128), 0, stream>>>(Hb, m_on_w + mi * 256, Hb, 4, 64, BS * 4);
    mlstm_combine<<<dim3(8192), dim3(256), 0, stream>>>(Hb, XC, UP, m_skip + mi * 256, Qb, BS * 256);
    gemm(Qb, m_down_W + (size_t)mi * 256 * 128, X, X, BS, 128, 256);
  };

  auto slstm = [&]() {
    layernorm128<<<dim3(BS), dim3(128), 0, stream>>>(X, s_ln_w, XN);
    conv_silu<<<dim3(4096), dim3(256), 0, stream>>>(XN, 128, s_conv_W, s_conv_b, XC, 128, BS * 128);
    slstm_gates<<<dim3(16384), dim3(256), 0, stream>>>(XC, XN, s_gates_W, s_bias, UP, BS * 512);
    slstm_scan<<<dim3(64), dim3(128), 0, stream>>>(UP, s_R, Kb);
    groupnorm_ws<<<dim3(BS), dim3(128), 0, stream>>>(Kb, s_gn_w, Kb, 4, 32, BS * 4);
    add_inplace<<<dim3(4096), dim3(256), 0, stream>>>(X, Kb, BS * 128);
    layernorm128<<<dim3(BS), dim3(128), 0, stream>>>(X, s_ffn_ln_w, XN);
    gemm(XN, s_ffn_up_W, nullptr, UP, BS, 384, 128);
    gelu_gate<<<dim3(6144), dim3(256), 0, stream>>>(UP, Vb, BS * 192);
    gemm(Vb, s_ffn_dn_W, X, X, BS, 128, 192);
  };

  int mi = 0;
  for (int blk = 0; blk < 7; ++blk) {
    if (blk == 1) slstm();
    else { mlstm(mi); ++mi; }
  }

  layernorm128<<<dim3(BS), dim3(128), 0, stream>>>(X, post_norm, XN);
  final_fc<<<dim3(16), dim3(128), 0, stream>>>(XN, fc_W, fc_b, (float*)d_out);
}